// TrainablePCEN_47545287967355
// MI455X (gfx1250) — compile-verified
//
#include <hip/hip_runtime.h>
#include <hip/hip_bf16.h>
#include <stdint.h>

// Problem constants from the reference: mel_spec [B=32, F=128, T=4096] f32.
#define ROWS   4096          // B*F independent IIR sequences
#define TLEN   4096          // time length
#define RANGE  128           // timesteps per wave-task in pass 3 / per carry range
#define NRANGE (TLEN / RANGE)  // 32
#define CHUNK  16
#define NCHUNK (RANGE / CHUNK) // 8
#define PCEN_EPS 1e-6f

typedef __attribute__((ext_vector_type(2))) float        v2f;
typedef __attribute__((ext_vector_type(8))) float        v8f;
typedef __attribute__((ext_vector_type(4))) unsigned int v4u;
typedef __attribute__((ext_vector_type(4))) int          v4i;
typedef __attribute__((ext_vector_type(8))) int          v8i;

// Exact integer power (safe at a==0: ipow(0,0)==1).
__device__ __forceinline__ float ipow(float a, int n) {
  float r = 1.0f, b = a;
  while (n) { if (n & 1) r *= b; b *= b; n >>= 1; }
  return r;
}

// Native-transcendental pow for strictly-positive bases:
// lowers to v_log_f32 + v_mul + v_exp_f32 (quarter-rate, ~3 ops) instead of
// the ~60-op precise __ocml_pow expansion that dominated the last build.
__device__ __forceinline__ float fpow(float x, float y) {
  return __builtin_amdgcn_exp2f(y * __builtin_amdgcn_logf(x));
}

// ---------------- TDM descriptor packing (CDNA5 ISA) ---
// Group 0: count=1 | lds_addr[63:32] | global_addr[120:64] | type=2 at [127:126]
__device__ __forceinline__ v4u tdm_g0(unsigned lds_off, const void* gptr) {
  unsigned long long ga = (unsigned long long)(uintptr_t)gptr;
  v4u g;
  g.x = 1u;                                   // count=1, user desc, no gather
  g.y = lds_off;                              // LDS byte address
  g.z = (unsigned)(ga & 0xFFFFFFFFull);       // global addr [31:0]
  g.w = (unsigned)((ga >> 32) & 0x01FFFFFFull) | (2u << 30); // addr[56:32], type=2
  return g;
}
// Group 1: data_size=4B; tensor_dim0/1, tile_dim0/1, tensor_dim0_stride (elements)
__device__ __forceinline__ v8i tdm_g1(unsigned td0, unsigned td1,
                                      unsigned tl0, unsigned tl1,
                                      unsigned long long stride0) {
  v8i g;
  g[0] = (int)(2u << 16);                                        // data_size=4B
  g[1] = (int)((td0 & 0xFFFFu) << 16);                           // dim0 lo16 @ [63:48]
  g[2] = (int)(((td0 >> 16) & 0xFFFFu) | ((td1 & 0xFFFFu) << 16));
  g[3] = (int)(((td1 >> 16) & 0xFFFFu) | ((tl0 & 0xFFFFu) << 16));
  g[4] = (int)(tl1 & 0xFFFFu);                                   // tile_dim1 (tile_dim2=0)
  g[5] = (int)(stride0 & 0xFFFFFFFFull);                         // dim0 stride lo32
  g[6] = (int)((stride0 >> 32) & 0xFFFFull);                     // dim0 stride hi16
  g[7] = 0;                                                      // dim1 stride unused (2D tile)
  return g;
}

__device__ __forceinline__ void tdm_load(v4u g0, v8i g1) {
  v4i z4 = (v4i)0;
#if __has_include(<hip/amd_detail/amd_gfx1250_TDM.h>)
  v8i z8 = (v8i)0;
  __builtin_amdgcn_tensor_load_to_lds(g0, g1, z4, z4, z8, 0);
#else
  __builtin_amdgcn_tensor_load_to_lds(g0, g1, z4, z4, 0);
#endif
}
__device__ __forceinline__ void tdm_store(v4u g0, v8i g1) {
  v4i z4 = (v4i)0;
#if __has_include(<hip/amd_detail/amd_gfx1250_TDM.h>)
  v8i z8 = (v8i)0;
  __builtin_amdgcn_tensor_store_from_lds(g0, g1, z4, z4, z8, 0);
#else
  __builtin_amdgcn_tensor_store_from_lds(g0, g1, z4, z4, 0);
#endif
}

// ---------------------------------------------------------------------------
// Pass 1+2 fused: one wave per row. Coalesced b128 loads; per-range tail
// reduction with shfl_xor; serial carry chain across the 32 ranges.
// carry_ws[row][rg] = M[rg*RANGE - 1]  (0 for rg==0).
// ---------------------------------------------------------------------------
__global__ __launch_bounds__(256) void pcen_carry_kernel(
    const float* __restrict__ x,
    const float* __restrict__ s_ptr,
    float* __restrict__ carry_ws)
{
  const int lane = threadIdx.x & 31;
  const int row  = (int)((blockIdx.x * blockDim.x + threadIdx.x) >> 5);

  const float s = fminf(fmaxf(s_ptr[0], 1e-4f), 1.0f);
  const float a = 1.0f - s;

  const float* xr = x + (size_t)row * TLEN;

  float w[4];
#pragma unroll
  for (int q = 0; q < 4; ++q) w[q] = ipow(a, 127 - 4 * lane - q);
  const float aR = ipow(a, RANGE);

  float carry = 0.0f;
  for (int rg = 0; rg < NRANGE; ++rg) {
    if (lane == 0) carry_ws[(size_t)row * NRANGE + rg] = carry;
    const float4 xv = *(const float4*)(xr + rg * RANGE + lane * 4);
    // y[0] = x[0] (coefficient 1), y[t>=1] = s*x[t]
    const float y0 = (rg == 0 && lane == 0) ? xv.x : s * xv.x;
    float p = w[0] * y0 + w[1] * (s * xv.y) + w[2] * (s * xv.z) + w[3] * (s * xv.w);
#pragma unroll
    for (int off = 16; off > 0; off >>= 1) p += __shfl_xor(p, off, 32);
    carry = aR * carry + p;   // uniform across the wave after the butterfly
  }
}

// ---------------------------------------------------------------------------
// Pass 3: one wave per (16-row group, 128-t range). TDM pulls the strided
// 16x128 f32 tile into LDS; per 16x16 chunk, the local scan is L(16x16) @
// Y(16x16) done as 4 chained V_WMMA_F32_16X16X4_F32 with the inter-chunk carry
// folded into the accumulator. PCEN nonlinearity applied in-place in LDS,
// result tile streamed back with TDM store.
// ---------------------------------------------------------------------------
__global__ __launch_bounds__(128) void pcen_main_kernel(
    const float* __restrict__ x,
    const float* __restrict__ s_ptr,
    const float* __restrict__ alpha_ptr,
    const float* __restrict__ delta_ptr,
    const float* __restrict__ r_ptr,
    const float* __restrict__ carry_ws,
    float* __restrict__ out)
{
  __shared__ float sh[4 * 16 * RANGE];   // 4 waves * 8KB = 32KB LDS

  const int lane = threadIdx.x & 31;
  const int wv   = threadIdx.x >> 5;
  const int task = (int)blockIdx.x * 4 + wv;       // 0 .. 8191
  const int rgp  = task >> 5;                      // rowgroup 0..255  (task / NRANGE)
  const int rng  = task & (NRANGE - 1);            // range    0..31
  const int rowbase = rgp * 16;
  const int t0      = rng * RANGE;

  const float s     = fminf(fmaxf(s_ptr[0], 1e-4f), 1.0f);
  const float a     = 1.0f - s;
  const float alpha = fminf(fmaxf(alpha_ptr[0], 0.0f), 1.0f);
  const float delta = fminf(fmaxf(delta_ptr[0], 0.0f), 50.0f);
  const float rr    = fminf(fmaxf(r_ptr[0], 0.0f), 1.0f);
  const float dr    = fpow(delta, rr);
  const float nalpha = -alpha;

  float* tile = sh + wv * (16 * RANGE);
  const unsigned lds_off = (unsigned)(uintptr_t)tile; // low 32 bits of generic ptr = LDS offset

  // DMA 16 rows x 128 t (row stride TLEN elements) -> contiguous LDS [row][t]
  const float* gsrc = x + (size_t)rowbase * TLEN + t0;
  tdm_load(tdm_g0(lds_off, gsrc), tdm_g1(TLEN, 16, RANGE, 16, TLEN));
  __builtin_amdgcn_s_wait_tensorcnt(0);
  __asm__ volatile("" ::: "memory");

  const int n  = lane & 15;   // row within group == N index == M(j) index for A
  const int hi = lane >> 4;

  // A operand (lower-triangular decay matrix) in WMMA 16x4 f32 A-layout:
  // lane holds M = lane&15; K = 2*(lane>>4) + v.  A[j][i] = a^(j-i), i<=j.
  v2f Amat[4];
#pragma unroll
  for (int kb = 0; kb < 4; ++kb) {
#pragma unroll
    for (int v = 0; v < 2; ++v) {
      const int i = kb * 4 + hi * 2 + v;
      Amat[kb][v] = (i <= n) ? ipow(a, n - i) : 0.0f;
    }
  }
  // a^(j+1) per held C/D element: VGPR g holds M = g + 8*hi.
  float apj[8];
#pragma unroll
  for (int g = 0; g < 8; ++g) apj[g] = ipow(a, g + 8 * hi + 1);

  float carr = carry_ws[(size_t)(rowbase + n) * NRANGE + rng];

  for (int c = 0; c < NCHUNK; ++c) {
    // Accumulator pre-loaded with the carry term: C[j][n] = a^(j+1) * carry[n]
    v8f acc;
#pragma unroll
    for (int g = 0; g < 8; ++g) acc[g] = apj[g] * carr;

#pragma unroll
    for (int kb = 0; kb < 4; ++kb) {
      // B operand: B[k][n] = y[t0 + c*16 + (kb*4+k)] of row n; lane holds
      // N = lane&15, K = 2*(lane>>4) + v (mirrors the A striping).
      v2f Bop;
#pragma unroll
      for (int v = 0; v < 2; ++v) {
        const int i  = kb * 4 + hi * 2 + v;
        const int tl = c * CHUNK + i;
        const float xv   = tile[n * RANGE + tl];
        const float coef = ((t0 + tl) == 0) ? 1.0f : s;   // y[0]=x[0]
        Bop[v] = coef * xv;
      }
      acc = __builtin_amdgcn_wmma_f32_16x16x4_f32(
          false, Amat[kb], false, Bop, (short)0, acc, false, false);
    }

    // Next-chunk carry = M at local t=15 = D[15][n], held in VGPR7 of lane 16+n.
    carr = __shfl(acc[7], 16 + n, 32);

    // PCEN nonlinearity via native v_log_f32/v_exp_f32, division-free:
    //   x/(M+eps)^alpha = x * exp2(-alpha * log2(M+eps))
    //   pcen = exp2(r * log2(x*... + delta)) - delta^r
    // Overwrite this chunk's x in LDS (all B reads of chunk c precede writes).
#pragma unroll
    for (int g = 0; g < 8; ++g) {
      const int tl = c * CHUNK + g + 8 * hi;
      const float xv = tile[n * RANGE + tl];
      const float m  = acc[g];
      const float inv_sm = __builtin_amdgcn_exp2f(
          nalpha * __builtin_amdgcn_logf(m + PCEN_EPS));
      const float u = __builtin_fmaf(xv, inv_sm, delta);
      const float p = __builtin_amdgcn_exp2f(rr * __builtin_amdgcn_logf(u)) - dr;
      tile[n * RANGE + tl] = p;
    }
  }

  // Make the LDS result tile visible to the TDM before the store descriptor.
  __asm__ volatile("s_wait_dscnt 0" ::: "memory");
  float* gdst = out + (size_t)rowbase * TLEN + t0;
  tdm_store(tdm_g0(lds_off, gdst), tdm_g1(TLEN, 16, RANGE, 16, TLEN));
  __builtin_amdgcn_s_wait_tensorcnt(0);
}

// ---------------------------------------------------------------------------
extern "C" void kernel_launch(void* const* d_in, const int* in_sizes, int n_in,
                              void* d_out, int out_size, void* d_ws, size_t ws_size,
                              hipStream_t stream) {
  (void)in_sizes; (void)n_in; (void)out_size; (void)ws_size;
  const float* x     = (const float*)d_in[0];
  const float* s_p   = (const float*)d_in[1];
  const float* al_p  = (const float*)d_in[2];
  const float* de_p  = (const float*)d_in[3];
  const float* r_p   = (const float*)d_in[4];
  float*       out   = (float*)d_out;
  float*       carry = (float*)d_ws;   // ROWS * NRANGE floats = 512 KB

  // Pass 1+2: 4096 rows, one wave each -> 512 blocks x 256 threads.
  pcen_carry_kernel<<<ROWS / 8, 256, 0, stream>>>(x, s_p, carry);

  // Pass 3: 256 rowgroups x 32 ranges = 8192 wave-tasks, 4 waves/block.
  pcen_main_kernel<<<(256 * NRANGE) / 4, 128, 0, stream>>>(
      x, s_p, al_p, de_p, r_p, carry, out);
}